// InitializationPolicy_4312147165291
// MI455X (gfx1250) — compile-verified
//
#include <hip/hip_runtime.h>
#include <math.h>

#define DIM_SOL 4096
#define DIM_CTX 4096
#define DIM_HID 8192
#define WROW    (DIM_CTX + DIM_SOL)   // row length of W = 8192 floats

typedef float v2f __attribute__((ext_vector_type(2)));
typedef float v8f __attribute__((ext_vector_type(8)));

// ------------------------- async-copy helpers (CDNA5) -------------------------
#if __has_builtin(__builtin_amdgcn_global_load_async_to_lds_b128)
#define HAVE_ASYNC 1
// Exact parameter type from hipcc diagnostic: int __vector(4), AS1 src / AS3 dst.
typedef int avec4i __attribute__((__vector_size__(4 * sizeof(int))));
typedef avec4i __attribute__((address_space(1)))* g_v4i_ptr;
typedef avec4i __attribute__((address_space(3)))* l_v4i_ptr;
__device__ __forceinline__ void async_cp16(float* lds_dst, const float* gsrc) {
  __builtin_amdgcn_global_load_async_to_lds_b128(
      (g_v4i_ptr)gsrc, (l_v4i_ptr)lds_dst, 0, 0);
}
#else
#define HAVE_ASYNC 0
__device__ __forceinline__ void async_cp16(float* lds_dst, const float* gsrc) {
  *(float4*)lds_dst = *(const float4*)gsrc;   // synchronous fallback
}
#endif

__device__ __forceinline__ void wait_async_le2() {
#if HAVE_ASYNC
#if __has_builtin(__builtin_amdgcn_s_wait_asynccnt)
  __builtin_amdgcn_s_wait_asynccnt(2);
#else
  asm volatile("s_wait_asynccnt 2" ::: "memory");
#endif
#endif
}
__device__ __forceinline__ void wait_async_le0() {
#if HAVE_ASYNC
#if __has_builtin(__builtin_amdgcn_s_wait_asynccnt)
  __builtin_amdgcn_s_wait_asynccnt(0);
#else
  asm volatile("s_wait_asynccnt 0" ::: "memory");
#endif
#endif
}

// ------------------------- Phase 1: a0 = c + W[:, :CTX] @ context ------------
#if __has_builtin(__builtin_amdgcn_wmma_f32_16x16x4_f32)
#define HAVE_WMMA_F32 1
#else
#define HAVE_WMMA_F32 0
#endif

__global__ __launch_bounds__(256) void a0_gemv_kernel(
    const float* __restrict__ W, const float* __restrict__ ctx,
    const float* __restrict__ c, float* __restrict__ a0) {
#if HAVE_WMMA_F32
  // One wave computes 16 rows of a0 using V_WMMA_F32_16X16X4_F32.
  // A(16x4): lanes 0-15 hold K=0(v0),K=1(v1); lanes 16-31 hold K=2(v0),K=3(v1).
  // B(4x16): only column N=0 nonzero -> context chunk lives in lanes 0 and 16.
  // D(16x16): column 0 -> vgpr m (lane 0) for m=0..7, vgpr m-8 (lane 16) for 8..15.
  __shared__ float sctx[DIM_CTX];   // 16 KB: whole context staged once
  const int tid = threadIdx.x;
  for (int k = tid; k < DIM_CTX; k += 256) sctx[k] = ctx[k];
  __syncthreads();

  const int wave = tid >> 5, lane = tid & 31;
  const int half = lane >> 4, l = lane & 15;
  const int r0 = (blockIdx.x * 8 + wave) * 16;          // 64 blocks * 8 waves * 16 rows = 8192
  const float* wp = W + (size_t)(r0 + l) * WROW + 2 * half;
  const float bmask = (l == 0) ? 1.0f : 0.0f;           // keep EXEC full for WMMA

  v8f acc = {};
#pragma unroll 4
  for (int k = 0; k < DIM_CTX; k += 4) {
    v2f av = *(const v2f*)(wp + k);                     // A: 2 K-values of this row
    v2f bv;
    bv.x = bmask * sctx[k + 2 * half];
    bv.y = bmask * sctx[k + 2 * half + 1];
    acc = __builtin_amdgcn_wmma_f32_16x16x4_f32(
        false, av, false, bv, (short)0, acc, false, false);
  }
  if (l == 0) {
    const int rb = r0 + half * 8;
#pragma unroll
    for (int j = 0; j < 8; ++j) a0[rb + j] = acc[j] + c[rb + j];
  }
#else
  const int row = blockIdx.x * 256 + threadIdx.x;
  if (row < DIM_HID) {
    const float* wp = W + (size_t)row * WROW;
    float acc = 0.0f;
    for (int k = 0; k < DIM_CTX; k += 4) {
      float4 wv = *(const float4*)(wp + k);
      float4 cv = *(const float4*)(ctx + k);
      acc += wv.x * cv.x + wv.y * cv.y + wv.z * cv.z + wv.w * cv.w;
    }
    a0[row] = acc + c[row];
  }
#endif
}

// ------------------------- Phase 2: sequential autoregressive scan -----------
// One workgroup (1024 threads = 32 waves, single WGP). Each thread owns 8
// hidden elements of `a`/`h` in registers. V row i+1 is async-prefetched into
// the other LDS buffer while step i computes; W column i+1 is prefetched into
// registers. Two split barriers per step.
__global__ __launch_bounds__(1024) void nade_scan_kernel(
    const float* __restrict__ W, const float* __restrict__ V,
    const float* __restrict__ b, const float* __restrict__ u,
    const float* __restrict__ a0, float* __restrict__ out) {
  __shared__ float vbuf[2][DIM_HID];   // 64 KB double buffer for V rows
  __shared__ float wpart[32];          // per-wave partial sums
  __shared__ float sbc;                // broadcast of the sampled bit

  const int tid  = threadIdx.x;
  const int lane = tid & 31;
  const int wave = tid >> 5;
  const int base = tid * 8;            // this thread's hidden slice [base, base+8)

  float a[8], h[8], wcol[8];
#pragma unroll
  for (int j = 0; j < 8; ++j) a[j] = a0[base + j];

  // warm-up prefetch: V row 0 (async -> LDS) and W sol-column 0 (-> regs)
  async_cp16(&vbuf[0][base],     V + base);
  async_cp16(&vbuf[0][base + 4], V + base + 4);
#pragma unroll
  for (int j = 0; j < 8; ++j) wcol[j] = W[(size_t)(base + j) * WROW + DIM_CTX];

  bool  dirty = true;
  float logp  = 0.0f;

  for (int i = 0; i < DIM_SOL; ++i) {
    const int cur = i & 1;

    // launch prefetch of next V row, then wait for the current one (wave-local:
    // each thread consumes exactly the bytes it async-loaded itself).
    if (i + 1 < DIM_SOL) {
      const float* vn = V + (size_t)(i + 1) * DIM_HID + base;
      async_cp16(&vbuf[cur ^ 1][base],     vn);
      async_cp16(&vbuf[cur ^ 1][base + 4], vn + 4);
      wait_async_le2();                 // the 2 outstanding are next-row loads
    } else {
      wait_async_le0();
    }

    if (dirty) {                        // sigmoid only when `a` changed
#pragma unroll
      for (int j = 0; j < 8; ++j) h[j] = 1.0f / (1.0f + expf(-a[j]));
      dirty = false;
    }

    // partial dot V_i . h over this thread's slice (b128 LDS reads)
    float4 v0 = *(const float4*)&vbuf[cur][base];
    float4 v1 = *(const float4*)&vbuf[cur][base + 4];
    float part = v0.x * h[0] + v0.y * h[1] + v0.z * h[2] + v0.w * h[3]
               + v1.x * h[4] + v1.y * h[5] + v1.z * h[6] + v1.w * h[7];
#pragma unroll
    for (int off = 16; off > 0; off >>= 1) part += __shfl_xor(part, off, 32);
    if (lane == 0) wpart[wave] = part;
    __syncthreads();

    if (wave == 0) {
      float t = wpart[lane];            // 32 wave partials, one per lane
#pragma unroll
      for (int off = 16; off > 0; off >>= 1) t += __shfl_xor(t, off, 32);
      if (lane == 0) {
        const float z  = b[i] + t;
        const float p  = 1.0f / (1.0f + expf(-z));
        const float sv = (u[i] < p) ? 1.0f : 0.0f;
        if (i < DIM_SOL - 2) {
          // numerically stable log sigmoid(+/-z)
          const float lsp = (z >= 0.0f) ? -log1pf(expf(-z)) : (z - log1pf(expf(z)));
          const float lsn = (z <= 0.0f) ? -log1pf(expf(z))  : (-z - log1pf(expf(-z)));
          logp += sv * lsp + (1.0f - sv) * lsn;
        }
        out[i] = sv;
        sbc    = sv;
      }
    }
    __syncthreads();

    if (sbc != 0.0f) {                  // autoregressive update a += W_col * s
#pragma unroll
      for (int j = 0; j < 8; ++j) a[j] += wcol[j];
      dirty = true;
    }
    if (i + 1 < DIM_SOL) {              // register-prefetch next W column
#pragma unroll
      for (int j = 0; j < 8; ++j)
        wcol[j] = W[(size_t)(base + j) * WROW + DIM_CTX + (i + 1)];
    }
  }

  if (tid == 0) out[DIM_SOL] = logp;    // tid 0 == wave 0 lane 0 (owns logp)
}

// ------------------------------- launcher ------------------------------------
extern "C" void kernel_launch(void* const* d_in, const int* in_sizes, int n_in,
                              void* d_out, int out_size, void* d_ws, size_t ws_size,
                              hipStream_t stream) {
  const float* ctx = (const float*)d_in[0];  // (4096)
  const float* u   = (const float*)d_in[1];  // (4096)
  const float* W   = (const float*)d_in[2];  // (8192, 8192)
  const float* V   = (const float*)d_in[3];  // (4096, 8192)
  const float* b   = (const float*)d_in[4];  // (4096)
  const float* c   = (const float*)d_in[5];  // (8192)
  float* out = (float*)d_out;                // 4096 samples + 1 logp
  float* a0  = (float*)d_ws;                 // 8192 floats scratch

  a0_gemv_kernel<<<dim3(64), dim3(256), 0, stream>>>(W, ctx, c, a0);
  nade_scan_kernel<<<dim3(1), dim3(1024), 0, stream>>>(W, V, b, u, a0, out);
}